// DataLoss_38525856645460
// MI455X (gfx1250) — compile-verified
//
#include <hip/hip_runtime.h>

typedef __attribute__((ext_vector_type(2))) float v2f;
typedef __attribute__((ext_vector_type(8))) float v8f;

// NaN-free min without fminf's canonicalization (v_max x,x) overhead.
__device__ __forceinline__ float min2f(float a, float b) {
    float d;
    asm("v_min_num_f32 %0, %1, %2" : "=v"(d) : "v"(a), "v"(b));
    return d;
}
__device__ __forceinline__ float min3f(float a, float b, float c) {
    float d;
    asm("v_min3_num_f32 %0, %1, %2, %3" : "=v"(d) : "v"(a), "v"(b), "v"(c));
    return d;
}

__device__ __forceinline__ v8f wmma_d2(v2f a, v2f b, v8f c) {
    return __builtin_amdgcn_wmma_f32_16x16x4_f32(
        false, a, false, b, (short)0, c, false, false);
}

// Kernel 1: one wave per (32-row template tile-pair, N-slice).
// D = A x B + C with A row m = (-2x,-2y,-2z,1), B col n = (x,y,z,||s||^2)^T,
// C = ||t_m||^2  =>  D[m,n] = ||t_m - s_n||^2 exactly.
// 2 M-tiles x 2 N-chunks per iteration: 4 WMMAs amortize 2 loads + 2 B-builds.
__global__ __launch_bounds__(32) void chamfer_tile_kernel(
    const float* __restrict__ scan,   // [N,3]
    const float* __restrict__ tmpl,   // [M,3]
    float* __restrict__ ws,           // [nslices, M]
    int M, int N, int nchunks, int cps)
{
    const int lane = threadIdx.x;        // full wave, EXEC all ones
    const int half = lane >> 4;          // 0: K={0,1}, 1: K={2,3}
    const int l16  = lane & 15;
    const int tb0  = blockIdx.x * 32;    // rows tb0..tb0+15 and tb0+16..tb0+31
    const int tb1  = tb0 + 16;
    const int slice = blockIdx.y;

    // ---- A and C operands for the two tiles ----
    v2f a0, a1; v8f c0v, c1v;
    {
        int m0 = tb0 + l16; if (m0 >= M) m0 = M - 1;
        int m1 = tb1 + l16; if (m1 >= M) m1 = M - 1;
        const float x0 = tmpl[3*m0], y0 = tmpl[3*m0+1], z0 = tmpl[3*m0+2];
        const float x1 = tmpl[3*m1], y1 = tmpl[3*m1+1], z1 = tmpl[3*m1+2];
        a0.x = half ? (-2.0f*z0) : (-2.0f*x0);  a0.y = half ? 1.0f : (-2.0f*y0);
        a1.x = half ? (-2.0f*z1) : (-2.0f*x1);  a1.y = half ? 1.0f : (-2.0f*y1);
        const float t0 = x0*x0 + y0*y0 + z0*z0;
        const float t1 = x1*x1 + y1*y1 + z1*z1;
        #pragma unroll
        for (int v = 0; v < 8; ++v) {
            c0v[v] = __shfl(t0, v + 8*half, 32);
            c1v[v] = __shfl(t1, v + 8*half, 32);
        }
    }

    float mn0[8], mn1[8];
    #pragma unroll
    for (int v = 0; v < 8; ++v) { mn0[v] = 3.402823466e38f; mn1[v] = 3.402823466e38f; }

    int ch  = slice * cps;
    int chE = ch + cps; if (chE > nchunks) chE = nchunks;
    int chF = N >> 4; if (chF > chE) chF = chE;     // end of full (unclamped) chunks

    const float* p = scan + (size_t)3 * ((size_t)ch * 16 + (size_t)l16);

    // ---- main loop: 2 chunks x 2 tiles per iteration ----
    for (; ch + 2 <= chF; ch += 2, p += 96) {
        __builtin_prefetch(p + 96, 0, 3);           // global_prefetch_b8, next pair
        const float sx0 = p[0],  sy0 = p[1],  sz0 = p[2];
        const float sx1 = p[48], sy1 = p[49], sz1 = p[50];
        const float q0 = sx0*sx0 + sy0*sy0 + sz0*sz0;
        const float q1 = sx1*sx1 + sy1*sy1 + sz1*sz1;
        v2f b0, b1;
        b0.x = half ? sz0 : sx0;  b0.y = half ? q0 : sy0;
        b1.x = half ? sz1 : sx1;  b1.y = half ? q1 : sy1;

        v8f d00 = wmma_d2(a0, b0, c0v);
        v8f d01 = wmma_d2(a0, b1, c0v);
        v8f d10 = wmma_d2(a1, b0, c1v);
        v8f d11 = wmma_d2(a1, b1, c1v);

        #pragma unroll
        for (int v = 0; v < 8; ++v) {
            mn0[v] = min3f(mn0[v], d00[v], d01[v]);
            mn1[v] = min3f(mn1[v], d10[v], d11[v]);
        }
    }

    // ---- leftover single full chunk ----
    for (; ch < chF; ++ch, p += 48) {
        const float sx = p[0], sy = p[1], sz = p[2];
        const float q = sx*sx + sy*sy + sz*sz;
        v2f b; b.x = half ? sz : sx; b.y = half ? q : sy;
        v8f d0 = wmma_d2(a0, b, c0v);
        v8f d1 = wmma_d2(a1, b, c1v);
        #pragma unroll
        for (int v = 0; v < 8; ++v) {
            mn0[v] = min2f(mn0[v], d0[v]);
            mn1[v] = min2f(mn1[v], d1[v]);
        }
    }

    // ---- partial tail chunk (only if N % 16 != 0), clamped columns ----
    for (; ch < chE; ++ch) {
        int n = ch * 16 + l16; if (n >= N) n = N - 1;
        const float* q3 = scan + (size_t)3 * (size_t)n;
        const float sx = q3[0], sy = q3[1], sz = q3[2];
        const float q = sx*sx + sy*sy + sz*sz;
        v2f b; b.x = half ? sz : sx; b.y = half ? q : sy;
        v8f d0 = wmma_d2(a0, b, c0v);
        v8f d1 = wmma_d2(a1, b, c1v);
        #pragma unroll
        for (int v = 0; v < 8; ++v) {
            mn0[v] = min2f(mn0[v], d0[v]);
            mn1[v] = min2f(mn1[v], d1[v]);
        }
    }

    // ---- min across the 16 lanes holding the same row, clamp >= 0 ----
    #pragma unroll
    for (int v = 0; v < 8; ++v) {
        float x = mn0[v];
        x = min2f(x, __shfl_xor(x, 1, 32));
        x = min2f(x, __shfl_xor(x, 2, 32));
        x = min2f(x, __shfl_xor(x, 4, 32));
        x = min2f(x, __shfl_xor(x, 8, 32));
        mn0[v] = fmaxf(x, 0.0f);
        float y = mn1[v];
        y = min2f(y, __shfl_xor(y, 1, 32));
        y = min2f(y, __shfl_xor(y, 2, 32));
        y = min2f(y, __shfl_xor(y, 4, 32));
        y = min2f(y, __shfl_xor(y, 8, 32));
        mn1[v] = fmaxf(y, 0.0f);
    }

    // lane 0 holds rows base..base+7, lane 16 holds rows base+8..base+15
    if (l16 == 0) {
        float* wrow = ws + (size_t)slice * M;
        const int r0 = tb0 + 8 * half;
        const int r1 = tb1 + 8 * half;
        #pragma unroll
        for (int v = 0; v < 8; ++v) {
            if (r0 + v < M) wrow[r0 + v] = mn0[v];
            if (r1 + v < M) wrow[r1 + v] = mn1[v];
        }
    }
}

// Kernel 2: deterministic final reduction — min over slices per row, sum rows.
__global__ __launch_bounds__(256) void chamfer_reduce_kernel(
    const float* __restrict__ ws, float* __restrict__ out, int M, int nsl)
{
    __shared__ float sm[256];
    float acc = 0.0f;
    for (int mIdx = (int)threadIdx.x; mIdx < M; mIdx += 256) {
        float v = ws[mIdx];
        for (int s = 1; s < nsl; ++s)
            v = min2f(v, ws[(size_t)s * M + mIdx]);
        acc += v;
    }
    sm[threadIdx.x] = acc;
    __syncthreads();
    for (int off = 128; off > 0; off >>= 1) {
        if ((int)threadIdx.x < off) sm[threadIdx.x] += sm[threadIdx.x + off];
        __syncthreads();
    }
    if (threadIdx.x == 0) out[0] = sm[0];
}

extern "C" void kernel_launch(void* const* d_in, const int* in_sizes, int n_in,
                              void* d_out, int out_size, void* d_ws, size_t ws_size,
                              hipStream_t stream) {
    const float* scan = (const float*)d_in[0];   // scan_vertices     [N,3]
    const float* tmpl = (const float*)d_in[1];   // template_vertices [M,3]
    float* out = (float*)d_out;                  // scalar f32
    float* ws  = (float*)d_ws;

    const int N = in_sizes[0] / 3;               // 20000
    const int M = in_sizes[1] / 3;               // 10000
    const int tilePairs = (M + 31) / 32;         // 313
    const int nchunks   = (N + 15) / 16;         // 1250

    // N-dimension slices for occupancy/latency hiding; cap by ws capacity.
    int nsl = (int)(ws_size / (sizeof(float) * (size_t)M));
    if (nsl < 1)  nsl = 1;
    if (nsl > 20) nsl = 20;
    const int cps = (nchunks + nsl - 1) / nsl;   // chunks per slice

    chamfer_tile_kernel<<<dim3(tilePairs, nsl), 32, 0, stream>>>(
        scan, tmpl, ws, M, N, nchunks, cps);
    chamfer_reduce_kernel<<<1, 256, 0, stream>>>(ws, out, M, nsl);
}